// LinOSSLayer_4930622455836
// MI455X (gfx1250) — compile-verified
//
#include <hip/hip_runtime.h>
#include <math.h>

// LinOSS layer for MI455X (gfx1250, wave32).
//   Phase 1: Fm_{re,im}[l,p] = step[p] * (X @ B_{re,im}^T)[l,p]    -- fp32 WMMA GEMM
//   Phase 2: chunked parallel linear scan (3 passes), async-to-LDS prefetch ring,
//            ys written in place over Fm
//   Phase 3: out = Ys_re@C_real^T - Ys_im@C_img^T + X*D            -- fp32 WMMA GEMM

typedef __attribute__((ext_vector_type(2))) float v2f;
typedef __attribute__((ext_vector_type(8))) float v8f;

#define L_SEQ  8192
#define H_DIM  128
#define P_DIM  256
#define CHUNK  128
#define NCHUNK (L_SEQ / CHUNK)   // 64
#define RING   8                 // async prefetch depth (rows)

#if defined(__gfx1250__) && __has_builtin(__builtin_amdgcn_global_load_async_to_lds_b32) && \
    __has_builtin(__builtin_amdgcn_s_wait_asynccnt)
#define USE_ASYNC 1
#else
#define USE_ASYNC 0
#endif

#if USE_ASYNC
// Builtin signature (from hipcc diagnostics): AS1 int* global src, AS3 int* LDS dst.
typedef __attribute__((address_space(1))) int gint;
typedef __attribute__((address_space(3))) int lint;
__device__ __forceinline__ void async_elem(const float* g, float* l) {
  // GLOBAL_LOAD_ASYNC_TO_LDS_B32: per-lane global -> LDS copy, tracked by ASYNCcnt.
  __builtin_amdgcn_global_load_async_to_lds_b32((gint*)g, (lint*)l, 0, 0);
}
#endif

__device__ __forceinline__ float sigmoidf_(float x) {
  return 1.0f / (1.0f + __expf(-x));
}

// ---------------------------------------------------------------------------
// Phase 1: Bu GEMM. One wave -> one 16x32 (l, p) tile: two p-subtiles for BOTH
// re and im accumulators (4x v8f), sharing the A load. Explicitly software-
// pipelined: next k-step's 5 b64 loads issued before this step's 4 WMMAs.
// ---------------------------------------------------------------------------
__global__ __launch_bounds__(128) void k_bu(const float* __restrict__ X,
                                            const float* __restrict__ Br,
                                            const float* __restrict__ Bi,
                                            const float* __restrict__ Sraw,
                                            float* __restrict__ Fre,
                                            float* __restrict__ Fim) {
  const int wave = blockIdx.x * 4 + (threadIdx.x >> 5);
  const int lane = threadIdx.x & 31;
  const int pairsP = P_DIM / 32;               // 8
  const int l0 = (wave / pairsP) * 16;
  const int p0 = (wave % pairsP) * 32;
  const int half = lane >> 4;                  // lanes 16-31: K+2 / M+8
  const int idx  = lane & 15;

  const float* __restrict__ xr  = X  + (size_t)(l0 + idx) * H_DIM + 2 * half;
  const float* __restrict__ b0r = Br + (size_t)(p0 + idx) * H_DIM + 2 * half;
  const float* __restrict__ b0i = Bi + (size_t)(p0 + idx) * H_DIM + 2 * half;
  const float* __restrict__ b1r = Br + (size_t)(p0 + 16 + idx) * H_DIM + 2 * half;
  const float* __restrict__ b1i = Bi + (size_t)(p0 + 16 + idx) * H_DIM + 2 * half;

  v8f c0r = {}, c0i = {}, c1r = {}, c1i = {};
  v2f a   = *(const v2f*)(xr);
  v2f v0r = *(const v2f*)(b0r);
  v2f v0i = *(const v2f*)(b0i);
  v2f v1r = *(const v2f*)(b1r);
  v2f v1i = *(const v2f*)(b1i);
#pragma unroll
  for (int k = 0; k < H_DIM - 4; k += 4) {
    v2f an  = *(const v2f*)(xr  + k + 4);
    v2f n0r = *(const v2f*)(b0r + k + 4);
    v2f n0i = *(const v2f*)(b0i + k + 4);
    v2f n1r = *(const v2f*)(b1r + k + 4);
    v2f n1i = *(const v2f*)(b1i + k + 4);
    c0r = __builtin_amdgcn_wmma_f32_16x16x4_f32(false, a, false, v0r, (short)0, c0r, false, false);
    c0i = __builtin_amdgcn_wmma_f32_16x16x4_f32(false, a, false, v0i, (short)0, c0i, false, false);
    c1r = __builtin_amdgcn_wmma_f32_16x16x4_f32(false, a, false, v1r, (short)0, c1r, false, false);
    c1i = __builtin_amdgcn_wmma_f32_16x16x4_f32(false, a, false, v1i, (short)0, c1i, false, false);
    a = an; v0r = n0r; v0i = n0i; v1r = n1r; v1i = n1i;
  }
  c0r = __builtin_amdgcn_wmma_f32_16x16x4_f32(false, a, false, v0r, (short)0, c0r, false, false);
  c0i = __builtin_amdgcn_wmma_f32_16x16x4_f32(false, a, false, v0i, (short)0, c0i, false, false);
  c1r = __builtin_amdgcn_wmma_f32_16x16x4_f32(false, a, false, v1r, (short)0, c1r, false, false);
  c1i = __builtin_amdgcn_wmma_f32_16x16x4_f32(false, a, false, v1i, (short)0, c1i, false, false);

  // Epilogue: fold the per-channel 'step' scale (Fm = step * Bu).
  const int   pA = p0 + idx;                   // C/D layout: N = lane&15
  const int   pB = p0 + 16 + idx;
  const float sA = sigmoidf_(Sraw[pA]);
  const float sB = sigmoidf_(Sraw[pB]);
#pragma unroll
  for (int i = 0; i < 8; ++i) {
    const int l = l0 + i + 8 * half;           // C/D layout: M = i + 8*(lane>=16)
    Fre[(size_t)l * P_DIM + pA] = sA * c0r[i];
    Fim[(size_t)l * P_DIM + pA] = sA * c0i[i];
    Fre[(size_t)l * P_DIM + pB] = sB * c1r[i];
    Fim[(size_t)l * P_DIM + pB] = sB * c1i[i];
  }
}

// ---------------------------------------------------------------------------
// Phase 2a: local chunk scans (zero init); store chunk-final (z,y).
// Recurrence per channel p (real coefficients, applied to re and im parts):
//   z' = z - s*a*y + f ;  y' = s*z + (1 - s^2*a)*y + f
// Async prefetch ring: 8 rows in flight via GLOBAL_LOAD_ASYNC_TO_LDS_B32.
// blockIdx.x encodes (chunk, part): part = bid & 1, chunk = bid >> 1.
// ---------------------------------------------------------------------------
__global__ __launch_bounds__(256) void k_scan_local(const float* __restrict__ Fre,
                                                    const float* __restrict__ Fim,
                                                    const float* __restrict__ Araw,
                                                    const float* __restrict__ Sraw,
                                                    float* __restrict__ bound) {
  __shared__ float ring[RING][P_DIM];          // 8 KB
  const int part = blockIdx.x & 1;
  const int c    = blockIdx.x >> 1;
  const int p    = threadIdx.x;
  const float a   = fmaxf(Araw[p], 0.0f);
  const float s   = sigmoidf_(Sraw[p]);
  const float m01 = -s * a;
  const float m11 = 1.0f - s * s * a;

  const float* __restrict__ F = (part ? Fim : Fre) + (size_t)c * CHUNK * P_DIM + p;

#if USE_ASYNC
#pragma unroll
  for (int i = 0; i < RING; ++i) async_elem(F + (size_t)i * P_DIM, &ring[i][p]);
#endif

  float z = 0.0f, y = 0.0f;
  for (int t = 0; t < CHUNK; ++t) {
#if USE_ASYNC
    __builtin_amdgcn_s_wait_asynccnt((short)(RING - 1));  // row t complete (in-order)
    const float f = ring[t & (RING - 1)][p];
#else
    const float f = F[(size_t)t * P_DIM];
#endif
    const float zn = z + m01 * y + f;
    const float yn = s * z + m11 * y + f;
    z = zn; y = yn;
#if USE_ASYNC
    const int tn = t + RING;
    if (tn < CHUNK) async_elem(F + (size_t)tn * P_DIM, &ring[t & (RING - 1)][p]);
#endif
  }
  float* b = bound + (((size_t)part * NCHUNK + c) * P_DIM + p) * 2;
  b[0] = z;
  b[1] = y;
}

// ---------------------------------------------------------------------------
// Phase 2b: sequential combine across the 64 chunk boundaries per (p, part).
// T = M^CHUNK via 7 squarings; stores the carry-IN state for each chunk.
// ---------------------------------------------------------------------------
__global__ __launch_bounds__(256) void k_scan_combine(const float* __restrict__ Araw,
                                                      const float* __restrict__ Sraw,
                                                      const float* __restrict__ bound,
                                                      float* __restrict__ carry) {
  const int part = blockIdx.x;
  const int p    = threadIdx.x;
  const float a = fmaxf(Araw[p], 0.0f);
  const float s = sigmoidf_(Sraw[p]);

  float t00 = 1.0f, t01 = -s * a, t10 = s, t11 = 1.0f - s * s * a;
#pragma unroll
  for (int i = 0; i < 7; ++i) {                // CHUNK = 128 = 2^7
    const float n00 = t00 * t00 + t01 * t10;
    const float n01 = t00 * t01 + t01 * t11;
    const float n10 = t10 * t00 + t11 * t10;
    const float n11 = t10 * t01 + t11 * t11;
    t00 = n00; t01 = n01; t10 = n10; t11 = n11;
  }

  float z = 0.0f, y = 0.0f;
  for (int c = 0; c < NCHUNK; ++c) {
    const size_t o = (((size_t)part * NCHUNK + c) * P_DIM + p) * 2;
    carry[o + 0] = z;
    carry[o + 1] = y;
    const float bz = bound[o + 0];
    const float by = bound[o + 1];
    const float zn = t00 * z + t01 * y + bz;
    const float yn = t10 * z + t11 * y + by;
    z = zn; y = yn;
  }
}

// ---------------------------------------------------------------------------
// Phase 2c: re-run each chunk from its true carry-in state, writing ys in
// place over Fm. In-place is safe: the async load of row t completed (enforced
// by s_wait_asynccnt) before row t's store is issued.
// ---------------------------------------------------------------------------
__global__ __launch_bounds__(256) void k_scan_apply(float* __restrict__ Fre,
                                                    float* __restrict__ Fim,
                                                    const float* __restrict__ Araw,
                                                    const float* __restrict__ Sraw,
                                                    const float* __restrict__ carry) {
  __shared__ float ring[RING][P_DIM];
  const int part = blockIdx.x & 1;
  const int c    = blockIdx.x >> 1;
  const int p    = threadIdx.x;
  const float a   = fmaxf(Araw[p], 0.0f);
  const float s   = sigmoidf_(Sraw[p]);
  const float m01 = -s * a;
  const float m11 = 1.0f - s * s * a;

  const size_t o = (((size_t)part * NCHUNK + c) * P_DIM + p) * 2;
  float z = carry[o + 0];
  float y = carry[o + 1];

  float* __restrict__ F = (part ? Fim : Fre) + (size_t)c * CHUNK * P_DIM + p;

#if USE_ASYNC
#pragma unroll
  for (int i = 0; i < RING; ++i) async_elem(F + (size_t)i * P_DIM, &ring[i][p]);
#endif

  for (int t = 0; t < CHUNK; ++t) {
#if USE_ASYNC
    __builtin_amdgcn_s_wait_asynccnt((short)(RING - 1));
    const float f = ring[t & (RING - 1)][p];
#else
    const float f = F[(size_t)t * P_DIM];
#endif
    const float zn = z + m01 * y + f;
    const float yn = s * z + m11 * y + f;
    z = zn; y = yn;
    F[(size_t)t * P_DIM] = yn;               // ys in place
#if USE_ASYNC
    const int tn = t + RING;
    if (tn < CHUNK) async_elem(F + (size_t)tn * P_DIM, &ring[t & (RING - 1)][p]);
#endif
  }
}

// ---------------------------------------------------------------------------
// Phase 3: out = Ys_re@C_real^T - Ys_im@C_img^T + X*D. One wave -> one 16x32
// (l, h) tile: two h-subtiles x (re, im) accumulators, pipelined like k_bu.
// ---------------------------------------------------------------------------
__global__ __launch_bounds__(128) void k_out(const float* __restrict__ Ysre,
                                             const float* __restrict__ Ysim,
                                             const float* __restrict__ Cr,
                                             const float* __restrict__ Ci,
                                             const float* __restrict__ X,
                                             const float* __restrict__ D,
                                             float* __restrict__ out) {
  const int wave = blockIdx.x * 4 + (threadIdx.x >> 5);
  const int lane = threadIdx.x & 31;
  const int pairsH = H_DIM / 32;               // 4
  const int l0 = (wave / pairsH) * 16;
  const int h0 = (wave % pairsH) * 32;
  const int half = lane >> 4;
  const int idx  = lane & 15;

  const float* __restrict__ yre = Ysre + (size_t)(l0 + idx) * P_DIM + 2 * half;
  const float* __restrict__ yim = Ysim + (size_t)(l0 + idx) * P_DIM + 2 * half;
  const float* __restrict__ c0r = Cr + (size_t)(h0 + idx) * P_DIM + 2 * half;       // (H,P): K-contig
  const float* __restrict__ c0i = Ci + (size_t)(h0 + idx) * P_DIM + 2 * half;
  const float* __restrict__ c1r = Cr + (size_t)(h0 + 16 + idx) * P_DIM + 2 * half;
  const float* __restrict__ c1i = Ci + (size_t)(h0 + 16 + idx) * P_DIM + 2 * half;

  v8f a0r = {}, a0i = {}, a1r = {}, a1i = {};
  v2f ar  = *(const v2f*)(yre);
  v2f ai  = *(const v2f*)(yim);
  v2f w0r = *(const v2f*)(c0r);
  v2f w0i = *(const v2f*)(c0i);
  v2f w1r = *(const v2f*)(c1r);
  v2f w1i = *(const v2f*)(c1i);
#pragma unroll
  for (int k = 0; k < P_DIM - 4; k += 4) {
    v2f arn = *(const v2f*)(yre + k + 4);
    v2f ain = *(const v2f*)(yim + k + 4);
    v2f n0r = *(const v2f*)(c0r + k + 4);
    v2f n0i = *(const v2f*)(c0i + k + 4);
    v2f n1r = *(const v2f*)(c1r + k + 4);
    v2f n1i = *(const v2f*)(c1i + k + 4);
    a0r = __builtin_amdgcn_wmma_f32_16x16x4_f32(false, ar, false, w0r, (short)0, a0r, false, false);
    a0i = __builtin_amdgcn_wmma_f32_16x16x4_f32(false, ai, false, w0i, (short)0, a0i, false, false);
    a1r = __builtin_amdgcn_wmma_f32_16x16x4_f32(false, ar, false, w1r, (short)0, a1r, false, false);
    a1i = __builtin_amdgcn_wmma_f32_16x16x4_f32(false, ai, false, w1i, (short)0, a1i, false, false);
    ar = arn; ai = ain; w0r = n0r; w0i = n0i; w1r = n1r; w1i = n1i;
  }
  a0r = __builtin_amdgcn_wmma_f32_16x16x4_f32(false, ar, false, w0r, (short)0, a0r, false, false);
  a0i = __builtin_amdgcn_wmma_f32_16x16x4_f32(false, ai, false, w0i, (short)0, a0i, false, false);
  a1r = __builtin_amdgcn_wmma_f32_16x16x4_f32(false, ar, false, w1r, (short)0, a1r, false, false);
  a1i = __builtin_amdgcn_wmma_f32_16x16x4_f32(false, ai, false, w1i, (short)0, a1i, false, false);

  const int   hA = h0 + idx;
  const int   hB = h0 + 16 + idx;
  const float dA = D[hA];
  const float dB = D[hB];
#pragma unroll
  for (int i = 0; i < 8; ++i) {
    const int l = l0 + i + 8 * half;
    out[(size_t)l * H_DIM + hA] = a0r[i] - a0i[i] + X[(size_t)l * H_DIM + hA] * dA;
    out[(size_t)l * H_DIM + hB] = a1r[i] - a1i[i] + X[(size_t)l * H_DIM + hB] * dB;
  }
}

// ---------------------------------------------------------------------------
extern "C" void kernel_launch(void* const* d_in, const int* in_sizes, int n_in,
                              void* d_out, int out_size, void* d_ws, size_t ws_size,
                              hipStream_t stream) {
  (void)in_sizes; (void)n_in; (void)out_size; (void)ws_size;
  const float* X    = (const float*)d_in[0];  // (L,H)
  const float* Araw = (const float*)d_in[1];  // (P,)
  const float* Br   = (const float*)d_in[2];  // (P,H)
  const float* Bi   = (const float*)d_in[3];  // (P,H)
  const float* Cr   = (const float*)d_in[4];  // (H,P)
  const float* Ci   = (const float*)d_in[5];  // (H,P)
  const float* D    = (const float*)d_in[6];  // (H,)
  const float* Sraw = (const float*)d_in[7];  // (P,)
  float* out = (float*)d_out;                 // (L,H)

  // Workspace layout (floats): Fre | Fim | bound | carry  (~16.5 MB total)
  float* Fre   = (float*)d_ws;
  float* Fim   = Fre + (size_t)L_SEQ * P_DIM;
  float* bound = Fim + (size_t)L_SEQ * P_DIM;
  float* carry = bound + (size_t)2 * NCHUNK * P_DIM * 2;

  // Phase 1: 4096 waves (512 l-tiles x 8 p-pairs), 4 waves per 128-thread block.
  k_bu<<<(L_SEQ / 16) * (P_DIM / 32) / 4, 128, 0, stream>>>(X, Br, Bi, Sraw, Fre, Fim);

  // Phase 2: chunked scan. 2*NCHUNK blocks of P threads (p across the block).
  k_scan_local<<<2 * NCHUNK, P_DIM, 0, stream>>>(Fre, Fim, Araw, Sraw, bound);
  k_scan_combine<<<2, P_DIM, 0, stream>>>(Araw, Sraw, bound, carry);
  k_scan_apply<<<2 * NCHUNK, P_DIM, 0, stream>>>(Fre, Fim, Araw, Sraw, carry);

  // Phase 3: 2048 waves (512 l-tiles x 4 h-pairs), 4 waves per block.
  k_out<<<(L_SEQ / 16) * (H_DIM / 32) / 4, 128, 0, stream>>>(Fre, Fim, Cr, Ci, X, D, out);
}